// ChamferLoss2D_48524540510941
// MI455X (gfx1250) — compile-verified
//
#include <hip/hip_runtime.h>

#define BB 4
#define NN 4096

typedef __attribute__((ext_vector_type(2))) float v2f;
typedef __attribute__((ext_vector_type(4))) float v4f;
typedef __attribute__((ext_vector_type(8))) float v8f;

// ---------------------------------------------------------------------------
// Prep: build augmented A-rows [x0, x1, |x|^2, 1] and B-cols [-2x0, -2x1, 1, |x|^2]
// for all 3 sets x 4 batches x 4096 points, and zero the 24 directed-sum slots.
// One WMMA_F32_16X16X4_F32 on these augmented operands produces a 16x16 tile of
// squared L2 distances directly.
// ---------------------------------------------------------------------------
__global__ __launch_bounds__(256) void chamfer_prep(
    const float* __restrict__ p1, const float* __restrict__ p2,
    const float* __restrict__ p3,
    float* __restrict__ Arow, float* __restrict__ Bcol,
    float* __restrict__ dsum)
{
    int idx = blockIdx.x * 256 + threadIdx.x;       // 0 .. 3*B*N-1
    if (blockIdx.x == 0 && threadIdx.x < 32) dsum[threadIdx.x] = 0.0f;
    if (idx >= 3 * BB * NN) return;

    int s = idx / (BB * NN);
    int r = idx - s * (BB * NN);                    // b*N + n
    const float* src = (s == 0) ? p1 : (s == 1) ? p2 : p3;

    float x0  = src[r * 2 + 0];
    float x1  = src[r * 2 + 1];
    float nrm = x0 * x0 + x1 * x1;

    v4f a  = { x0, x1, nrm, 1.0f };
    v4f bt = { -2.0f * x0, -2.0f * x1, 1.0f, nrm };
    *(v4f*)(Arow + (size_t)idx * 4) = a;
    *(v4f*)(Bcol + (size_t)idx * 4) = bt;
}

// ---------------------------------------------------------------------------
// Main: one directed chamfer pass per (dir, batch, row-tile-group).
//   dir 0..5 = (0,1),(1,0),(0,2),(2,0),(1,2),(2,1)   (rows-set, cols-set)
// Block = 256 threads = 8 wave32; block stages the whole augmented-B set for
// its (set,batch) into LDS (64 KB) via GLOBAL_LOAD_ASYNC_TO_LDS_B128 (ASYNCcnt,
// no VGPR round-trip), each wave owns one 16-row tile and sweeps all 256 column
// tiles with V_WMMA_F32_16X16X4_F32, accumulating an elementwise min of squared
// distances. sqrt/clamp happen once at the end (min commutes with sqrt).
// ---------------------------------------------------------------------------
__global__ __launch_bounds__(256) void chamfer_main(
    const float* __restrict__ Arow, const float* __restrict__ Bcol,
    float* __restrict__ dsum)
{
    __shared__ float ldsB[NN * 4];                  // 64 KB of a 320 KB WGP LDS

    const int xs[6] = {0, 1, 0, 2, 1, 2};
    const int ys[6] = {1, 0, 2, 0, 2, 1};

    int rtg = blockIdx.x & 31;                      // row-tile group (8 tiles)
    int b   = (blockIdx.x >> 5) & 3;                // batch
    int dir = blockIdx.x >> 7;                      // directed pair

    // Async stage of augmented B for (set=ys[dir], batch=b): 4096 x 16B -> LDS.
    // Each thread issues 16 async b128 copies; memory -> LDS directly, tracked
    // by ASYNCcnt (no VGPR staging).
    const float* Bsrc = Bcol + ((size_t)(ys[dir] * BB + b)) * NN * 4;
    for (int i = threadIdx.x; i < NN; i += 256) {
        unsigned lds_off   = (unsigned)(uintptr_t)(&ldsB[i * 4]); // flat low 32 = LDS byte addr
        unsigned long long gaddr = (unsigned long long)(uintptr_t)(Bsrc + (size_t)i * 4);
        asm volatile("global_load_async_to_lds_b128 %0, %1, off"
                     :: "v"(lds_off), "v"(gaddr) : "memory");
    }
    asm volatile("s_wait_asynccnt 0" ::: "memory");
    __syncthreads();

    int wave = threadIdx.x >> 5;
    int lane = threadIdx.x & 31;
    int l16  = lane & 15;
    int koff = (lane < 16) ? 0 : 2;                 // K=0,1 in low half; K=2,3 in high half

    // A fragment (32-bit A 16x4 layout): lane holds row M=l16, 2 K-values.
    int row = (rtg * 8 + wave) * 16 + l16;
    const float* Asrc = Arow + ((size_t)(xs[dir] * BB + b)) * NN * 4;
    v2f afrag = *(const v2f*)(Asrc + (size_t)row * 4 + koff);

    v8f acc;
#pragma unroll
    for (int i = 0; i < 8; ++i) acc[i] = 3.0e38f;
    v8f czero = {};

    for (int ct = 0; ct < NN / 16; ++ct) {
        // B fragment (32-bit B 4x16 layout): lane holds col N=l16, rows K per half.
        v2f bfrag = *(const v2f*)(&ldsB[(ct * 16 + l16) * 4 + koff]);
        // D = A x B + 0  ->  16x16 tile of squared distances
        v8f d = __builtin_amdgcn_wmma_f32_16x16x4_f32(
            /*neg_a=*/false, afrag, /*neg_b=*/false, bfrag,
            /*c_mod=*/(short)0, czero, /*reuse_a=*/false, /*reuse_b=*/false);
#pragma unroll
        for (int i = 0; i < 8; ++i) acc[i] = __builtin_fminf(acc[i], d[i]);
    }

    // Row-min over the 16 column slots (xor-shuffles stay within each 16-lane
    // half), then sqrt(max(.,0)) and sum the 8 rows this half-wave owns.
    float sum8 = 0.0f;
#pragma unroll
    for (int i = 0; i < 8; ++i) {
        float v = acc[i];
#pragma unroll
        for (int off = 1; off < 16; off <<= 1)
            v = __builtin_fminf(v, __shfl_xor(v, off, 32));
        sum8 += __builtin_sqrtf(__builtin_fmaxf(v, 0.0f));
    }
    // lanes 0-15 summed rows 0..7, lanes 16-31 summed rows 8..15
    sum8 += __shfl_xor(sum8, 16, 32);
    if (lane == 0) atomicAdd(dsum + dir * BB + b, sum8);
}

// ---------------------------------------------------------------------------
// Finalize: out[b] = mean over 3 pairs of (MARGIN - LW * chamfer(pair))
//   chamfer = (mean_row_min_dir + mean_row_min_revdir) / 2
// ---------------------------------------------------------------------------
__global__ void chamfer_final(const float* __restrict__ dsum, float* __restrict__ out)
{
    int b = threadIdx.x;
    if (b >= BB) return;
    const float invN = 1.0f / (float)NN;
    float ch = 0.0f;
#pragma unroll
    for (int p = 0; p < 3; ++p) {
        float d1 = dsum[(2 * p + 0) * BB + b] * invN;
        float d2 = dsum[(2 * p + 1) * BB + b] * invN;
        ch += 0.5f * (d1 + d2);
    }
    // (sum_p (MARGIN - LW*ch_p)) / 3  with MARGIN = LOSS_WEIGHT = 1
    out[b] = (3.0f - ch) / 3.0f;
}

extern "C" void kernel_launch(void* const* d_in, const int* in_sizes, int n_in,
                              void* d_out, int out_size, void* d_ws, size_t ws_size,
                              hipStream_t stream)
{
    const float* p1 = (const float*)d_in[0];
    const float* p2 = (const float*)d_in[1];
    const float* p3 = (const float*)d_in[2];

    float* Arow = (float*)d_ws;                         // 3*4*4096*4 floats = 768 KB
    float* Bcol = Arow + (size_t)3 * BB * NN * 4;       // 768 KB
    float* dsum = Bcol + (size_t)3 * BB * NN * 4;       // 32 floats (24 used)

    int prep_blocks = (3 * BB * NN + 255) / 256;        // 192
    chamfer_prep<<<prep_blocks, 256, 0, stream>>>(p1, p2, p3, Arow, Bcol, dsum);

    chamfer_main<<<6 * BB * 32, 256, 0, stream>>>(Arow, Bcol, dsum);  // 768 blocks

    chamfer_final<<<1, 32, 0, stream>>>(dsum, (float*)d_out);
}